// StreampTFGridNet_Small_18683107737990
// MI455X (gfx1250) — compile-verified
//
#include <hip/hip_runtime.h>
#include <math.h>

typedef float v2f __attribute__((ext_vector_type(2)));
typedef float v8f __attribute__((ext_vector_type(8)));

#define EPSF 1e-5f

__device__ __forceinline__ float sigm(float x) { return 1.f / (1.f + expf(-x)); }

// ---------------------------------------------------------------------------
// Generic fp32 WMMA GEMM:  C[M,N] = A[M,K] * B (+bias[N])
//   BT=1: B stored (N,K) row-major (C = A * B^T, the "x @ W.T" case)
//   BT=0: B stored (K,N) row-major
// K % 4 == 0 required. Each wave computes a 16(M) x 64(N) strip with 4
// v_wmma_f32_16x16x4_f32 accumulators (A fragment reused 4x). 8 waves/block
// arranged 4(M) x 2(N) cover a 64 x 128 block tile. Branch-free inner loop.
// ---------------------------------------------------------------------------
template <int BT>
__global__ __launch_bounds__(256) void k_gemm_t(const float* __restrict__ A,
                                                const float* __restrict__ Bm,
                                                const float* __restrict__ bias,
                                                float* __restrict__ Cm,
                                                int M, int N, int K) {
  const int lane = threadIdx.x & 31;
  const int wave = threadIdx.x >> 5;
  const int wm = wave >> 1;  // 0..3
  const int wn = wave & 1;   // 0..1
  const int m0 = blockIdx.y * 64 + wm * 16;
  const int n0 = blockIdx.x * 128 + wn * 64;
  const int mi = lane & 15;
  const int half = lane >> 4;  // selects K pair {0,1} vs {2,3}
  const int arow = (m0 + mi < M) ? (m0 + mi) : (M - 1);
  int bcol[4];
#pragma unroll
  for (int tb = 0; tb < 4; ++tb) {
    int c = n0 + tb * 16 + mi;
    bcol[tb] = (c < N) ? c : (N - 1);
  }
  v8f acc[4];
#pragma unroll
  for (int tb = 0; tb < 4; ++tb) acc[tb] = (v8f){0.f, 0.f, 0.f, 0.f, 0.f, 0.f, 0.f, 0.f};
  const float* __restrict__ Ar = A + (long)arow * K;
  for (int k0 = 0; k0 < K; k0 += 4) {
    const int ka = k0 + half * 2;
    v2f a;
    a.x = Ar[ka];
    a.y = Ar[ka + 1];
    v2f b[4];
#pragma unroll
    for (int tb = 0; tb < 4; ++tb) {
      if (BT) {
        b[tb].x = Bm[(long)bcol[tb] * K + ka];
        b[tb].y = Bm[(long)bcol[tb] * K + ka + 1];
      } else {
        b[tb].x = Bm[(long)ka * N + bcol[tb]];
        b[tb].y = Bm[(long)(ka + 1) * N + bcol[tb]];
      }
    }
#pragma unroll
    for (int tb = 0; tb < 4; ++tb)
      acc[tb] = __builtin_amdgcn_wmma_f32_16x16x4_f32(false, a, false, b[tb],
                                                      (short)0, acc[tb], false, false);
  }
#pragma unroll
  for (int tb = 0; tb < 4; ++tb) {
    const int cn = n0 + tb * 16 + mi;
    if (cn < N) {
      const float bb = bias ? bias[cn] : 0.f;
#pragma unroll
      for (int i = 0; i < 8; ++i) {
        const int cm = m0 + half * 8 + i;
        if (cm < M) Cm[(long)cm * N + cn] = acc[tb][i] + bb;
      }
    }
  }
}

// ---------------------------------------------------------------------------
// Fused GRU scan step: gh = h * Whh^T + bhh (WMMA, 16 rows x 576 cols per
// block, 9 waves each doing a 16x64 strip), gh staged in LDS, then the GRU
// gate pointwise + h/y update in the same block. One launch per scan step.
// gridDim.x = Nrows/16 (Nrows % 16 == 0).
// ---------------------------------------------------------------------------
__global__ __launch_bounds__(288) void k_gru_fused(
    float* __restrict__ h, const float* __restrict__ whh,
    const float* __restrict__ bhh, const float* __restrict__ xp,
    float* __restrict__ y, int seqLen, int step,
    int ySeqLen, int yStep, int yRowStride, int yOff) {
  __shared__ float ghL[16][576];
  const int lane = threadIdx.x & 31;
  const int wave = threadIdx.x >> 5;  // 0..8, covers n0 = wave*64
  const int n0 = wave * 64;
  const int mi = lane & 15;
  const int half = lane >> 4;
  const long rowBase = (long)blockIdx.x * 16;
  const long arow = rowBase + mi;
  v8f acc[4];
#pragma unroll
  for (int tb = 0; tb < 4; ++tb) acc[tb] = (v8f){0.f, 0.f, 0.f, 0.f, 0.f, 0.f, 0.f, 0.f};
  const float* __restrict__ Ar = h + arow * 192;
  for (int k0 = 0; k0 < 192; k0 += 4) {
    const int ka = k0 + half * 2;
    v2f a;
    a.x = Ar[ka];
    a.y = Ar[ka + 1];
    v2f b[4];
#pragma unroll
    for (int tb = 0; tb < 4; ++tb) {
      const long bc = n0 + tb * 16 + mi;
      b[tb].x = whh[bc * 192 + ka];
      b[tb].y = whh[bc * 192 + ka + 1];
    }
#pragma unroll
    for (int tb = 0; tb < 4; ++tb)
      acc[tb] = __builtin_amdgcn_wmma_f32_16x16x4_f32(false, a, false, b[tb],
                                                      (short)0, acc[tb], false, false);
  }
#pragma unroll
  for (int tb = 0; tb < 4; ++tb) {
    const int cn = n0 + tb * 16 + mi;
    const float bb = bhh[cn];
#pragma unroll
    for (int i = 0; i < 8; ++i) ghL[half * 8 + i][cn] = acc[tb][i] + bb;
  }
  __syncthreads();
  for (int idx = threadIdx.x; idx < 16 * 192; idx += 288) {
    const int rr = idx / 192, j = idx % 192;
    const long gn = rowBase + rr;
    const float* xr = xp + (gn * seqLen + step) * 576;
    float rg = sigm(xr[j] + ghL[rr][j]);
    float zg = sigm(xr[192 + j] + ghL[rr][192 + j]);
    float ng = tanhf(xr[384 + j] + rg * ghL[rr][384 + j]);
    float hv = h[gn * 192 + j];
    float hn = (1.f - zg) * ng + zg * hv;
    h[gn * 192 + j] = hn;
    y[(gn * ySeqLen + yStep) * yRowStride + yOff + j] = hn;
  }
}

// ---------------------------------------------------------------------------
// Utility kernels
// ---------------------------------------------------------------------------
__global__ void k_zero(float* p, int n) {
  int i = blockIdx.x * blockDim.x + threadIdx.x;
  if (i < n) p[i] = 0.f;
}
__global__ void k_copy(const float* __restrict__ s, float* __restrict__ d, int n) {
  int i = blockIdx.x * blockDim.x + threadIdx.x;
  if (i < n) d[i] = s[i];
}

// ---------------------------------------------------------------------------
// Encoder: x = concat(cache(2 frames), mix(4 frames)) in time; 3x3 conv
// (ph=0 -> Tout=4, pw=1 -> Fout=129), + bias, + batchnorm.  Out: (B,32,4,129)
// ---------------------------------------------------------------------------
__global__ void k_enc(const float* __restrict__ mix, const float* __restrict__ cache,
                      const float* __restrict__ w, const float* __restrict__ bvec,
                      const float* __restrict__ bnw, const float* __restrict__ bnb,
                      const float* __restrict__ bnm, const float* __restrict__ bnv,
                      float* __restrict__ esti) {
  int idx = blockIdx.x * blockDim.x + threadIdx.x;
  const int total = 64 * 32 * 4 * 129;
  if (idx >= total) return;
  int f = idx % 129; int r = idx / 129;
  int t = r % 4; r /= 4;
  int o = r % 32; int b = r / 32;
  float acc = bvec[o];
#pragma unroll
  for (int i = 0; i < 2; ++i)
#pragma unroll
    for (int kh = 0; kh < 3; ++kh)
#pragma unroll
      for (int kw = 0; kw < 3; ++kw) {
        int tt = t + kh;
        int ff = f + kw - 1;
        if (ff < 0 || ff >= 129) continue;
        float x = (tt < 2) ? cache[((b * 2 + i) * 2 + tt) * 129 + ff]
                           : mix[((b * 2 + i) * 4 + (tt - 2)) * 129 + ff];
        acc += x * w[((o * 2 + i) * 3 + kh) * 3 + kw];
      }
  acc = (acc - bnm[o]) * rsqrtf(bnv[o] + EPSF) * bnw[o] + bnb[o];
  esti[idx] = acc;
}

// enc_cache_new = mix[:, :, 2:4, :]
__global__ void k_enc_cache(const float* __restrict__ mix, float* __restrict__ dst) {
  int idx = blockIdx.x * blockDim.x + threadIdx.x;
  const int total = 64 * 2 * 2 * 129;
  if (idx >= total) return;
  int f = idx % 129; int r = idx / 129;
  int tp = r % 2; r /= 2;
  int i = r % 2; int b = r / 2;
  dst[idx] = mix[((b * 2 + i) * 4 + (2 + tp)) * 129 + f];
}

// ---------------------------------------------------------------------------
// Fusion MHA: one block per (b,f); 32 lanes = channels. T=4, heads=4, d=8.
// Writes auxn in (B,32,4,129) layout.
// ---------------------------------------------------------------------------
__global__ __launch_bounds__(32) void k_fusion_mha(
    const float* __restrict__ esti, const float* __restrict__ aux,
    const float* __restrict__ in_w, const float* __restrict__ in_b,
    const float* __restrict__ out_w, const float* __restrict__ out_b,
    float* __restrict__ auxn) {
  __shared__ float kL[4][32], vL[4][32], qL[32], attL[4][4][4], oL[4][32];
  int c = threadIdx.x;
  int b = blockIdx.x / 129, f = blockIdx.x % 129;
  const float* wq = in_w;         const float* bq = in_b;
  const float* wk = in_w + 1024;  const float* bk = in_b + 32;
  const float* wv = in_w + 2048;  const float* bv = in_b + 64;
  float q = bq[c];
  for (int c2 = 0; c2 < 32; ++c2) q += wq[c * 32 + c2] * aux[b * 32 + c2];
  qL[c] = q;
  for (int t = 0; t < 4; ++t) {
    float kk = bk[c], vv = bv[c];
    for (int c2 = 0; c2 < 32; ++c2) {
      float e = esti[((b * 32 + c2) * 4 + t) * 129 + f];
      kk += wk[c * 32 + c2] * e;
      vv += wv[c * 32 + c2] * e;
    }
    kL[t][c] = kk; vL[t][c] = vv;
  }
  __syncthreads();
  if (c < 16) {
    int h = c >> 2, t = c & 3;
    float sc[4], mx = -1e30f;
    for (int s = 0; s < 4; ++s) {
      float a = 0.f;
      for (int j = 0; j < 8; ++j) a += qL[h * 8 + j] * kL[s][h * 8 + j];
      sc[s] = a * rsqrtf(8.f);
      mx = fmaxf(mx, sc[s]);
    }
    float den = 0.f;
    for (int s = 0; s < 4; ++s) { sc[s] = expf(sc[s] - mx); den += sc[s]; }
    for (int s = 0; s < 4; ++s) attL[h][t][s] = sc[s] / den;
  }
  __syncthreads();
  int h = c >> 3;
  for (int t = 0; t < 4; ++t) {
    float o = 0.f;
    for (int s = 0; s < 4; ++s) o += attL[h][t][s] * vL[s][c];
    oL[t][c] = o;
  }
  __syncthreads();
  for (int t = 0; t < 4; ++t) {
    float op = out_b[c];
    for (int c2 = 0; c2 < 32; ++c2) op += out_w[c * 32 + c2] * oL[t][c2];
    auxn[((b * 32 + c) * 4 + t) * 129 + f] = qL[c] + op;
  }
}

// 1x1 conv over concat([esti, auxn]) channels (64 -> 32)
__global__ void k_fusion_conv(const float* __restrict__ esti,
                              const float* __restrict__ auxn,
                              const float* __restrict__ cvw,
                              const float* __restrict__ cvb,
                              float* __restrict__ out) {
  int idx = blockIdx.x * blockDim.x + threadIdx.x;
  const int total = 64 * 32 * 4 * 129;
  if (idx >= total) return;
  int f = idx % 129; int r = idx / 129;
  int t = r % 4; r /= 4;
  int o = r % 32; int b = r / 32;
  float acc = cvb[o];
  for (int c = 0; c < 32; ++c) {
    int xi = ((b * 32 + c) * 4 + t) * 129 + f;
    acc += cvw[o * 64 + c] * esti[xi];
    acc += cvw[o * 64 + 32 + c] * auxn[xi];
  }
  out[idx] = acc;
}

// ---------------------------------------------------------------------------
// LayerNorm over channels (axis 1) of (B,32,4,129); output same layout.
// ---------------------------------------------------------------------------
__global__ void k_ln4d(const float* __restrict__ x, const float* __restrict__ g,
                       const float* __restrict__ bb, float* __restrict__ y) {
  int idx = blockIdx.x * blockDim.x + threadIdx.x;
  const int total = 64 * 4 * 129;
  if (idx >= total) return;
  int f = idx % 129; int r = idx / 129;
  int t = r % 4; int b = r / 4;
  long base = ((long)b * 32 * 4 + t) * 129 + f;
  float s1 = 0.f, s2 = 0.f;
  for (int c = 0; c < 32; ++c) {
    float v = x[base + (long)c * 516];
    s1 += v; s2 += v * v;
  }
  float mu = s1 / 32.f;
  float var = s2 / 32.f - mu * mu;
  float inv = rsqrtf(var + EPSF);
  for (int c = 0; c < 32; ++c) {
    float v = x[base + (long)c * 516];
    y[base + (long)c * 516] = (v - mu) * inv * g[c] + bb[c];
  }
}

// LayerNorm over channels of intra, emitted transposed as xn (B*F, T, C)
__global__ void k_ln4d_to_xn(const float* __restrict__ x, const float* __restrict__ g,
                             const float* __restrict__ bb, float* __restrict__ xn) {
  int idx = blockIdx.x * blockDim.x + threadIdx.x;
  const int total = 64 * 4 * 129;
  if (idx >= total) return;
  int f = idx % 129; int r = idx / 129;
  int t = r % 4; int b = r / 4;
  long base = ((long)b * 32 * 4 + t) * 129 + f;
  float s1 = 0.f, s2 = 0.f;
  for (int c = 0; c < 32; ++c) {
    float v = x[base + (long)c * 516];
    s1 += v; s2 += v * v;
  }
  float mu = s1 / 32.f;
  float var = s2 / 32.f - mu * mu;
  float inv = rsqrtf(var + EPSF);
  long ob = (((long)b * 129 + f) * 4 + t) * 32;
  for (int c = 0; c < 32; ++c) {
    float v = x[base + (long)c * 516];
    xn[ob + c] = (v - mu) * inv * g[c] + bb[c];
  }
}

// gin[(b*4+t), l, c*8+k] = xi[b,c,t,l+k]   (unfold along F)
__global__ void k_build_gin(const float* __restrict__ xi, float* __restrict__ gin) {
  int idx = blockIdx.x * blockDim.x + threadIdx.x;
  const int total = 256 * 122 * 256;
  if (idx >= total) return;
  int q = idx % 256;
  int l = (idx / 256) % 122;
  int n = idx / (256 * 122);
  int c = q >> 3, k = q & 7;
  int b = n >> 2, t = n & 3;
  gin[idx] = xi[((b * 32 + c) * 4 + t) * 129 + (l + k)];
}

// overlap-add of contrib (B*T rows, cols c*8+k over window l) + lb + residual x
__global__ void k_unfold(const float* __restrict__ contrib, const float* __restrict__ x,
                         const float* __restrict__ lb, float* __restrict__ intra) {
  int idx = blockIdx.x * blockDim.x + threadIdx.x;
  const int total = 256 * 32 * 129;
  if (idx >= total) return;
  int f = idx % 129; int r = idx / 129;
  int c = r % 32; int n = r / 32;
  float acc = lb[c];
#pragma unroll
  for (int k = 0; k < 8; ++k) {
    int l = f - k;
    if (l >= 0 && l < 122) acc += contrib[((long)n * 122 + l) * 256 + c * 8 + k];
  }
  int b = n >> 2, t = n & 3;
  long xi = ((long)(b * 32 + c) * 4 + t) * 129 + f;
  intra[xi] = acc + x[xi];
}

// inter[b,c,t,f] = lin[(b*F+f)*T+t, c] + intra[b,c,t,f]
__global__ void k_add_lin(const float* __restrict__ intra, const float* __restrict__ lin,
                          float* __restrict__ inter) {
  int idx = blockIdx.x * blockDim.x + threadIdx.x;
  const int total = 64 * 32 * 4 * 129;
  if (idx >= total) return;
  int f = idx % 129; int r = idx / 129;
  int t = r % 4; r /= 4;
  int c = r % 32; int b = r / 32;
  inter[idx] = intra[idx] + lin[(((long)b * 129 + f) * 4 + t) * 32 + c];
}

// ---------------------------------------------------------------------------
// QKV make: one block per (j,h,b,t). 1x1 proj (32->4) + PReLU + ln4dcf over
// (E,F)=516 values, writes Q (NHB,T,516), Kt (NHB,516,T), V (NHB,T,516).
// ---------------------------------------------------------------------------
__global__ __launch_bounds__(128) void k_qkv_make(
    const float* __restrict__ inter, const float* __restrict__ qw,
    const float* __restrict__ qb, const float* __restrict__ qa,
    const float* __restrict__ qlg, const float* __restrict__ qlb,
    float* __restrict__ Qb, float* __restrict__ Ktb, float* __restrict__ Vb) {
  __shared__ float val[516];
  __shared__ float red1[128], red2[128];
  int blk = blockIdx.x;
  int t = blk & 3; blk >>= 2;
  int b = blk & 63; blk >>= 6;
  int h = blk & 3; int j = blk >> 2;
  const float* w = qw + (j * 4 + h) * 128;
  const float* bias = qb + (j * 4 + h) * 4;
  float a = qa[j * 4 + h];
  const float* g = qlg + (j * 4 + h) * 516;
  const float* gb = qlb + (j * 4 + h) * 516;
  float s1 = 0.f, s2 = 0.f;
  for (int idx = threadIdx.x; idx < 516; idx += 128) {
    int e = idx / 129, f = idx % 129;
    float v = bias[e];
    for (int c = 0; c < 32; ++c)
      v += w[e * 32 + c] * inter[((b * 32 + c) * 4 + t) * 129 + f];
    v = (v >= 0.f) ? v : a * v;
    val[idx] = v; s1 += v; s2 += v * v;
  }
  red1[threadIdx.x] = s1; red2[threadIdx.x] = s2;
  __syncthreads();
  for (int s = 64; s > 0; s >>= 1) {
    if (threadIdx.x < s) {
      red1[threadIdx.x] += red1[threadIdx.x + s];
      red2[threadIdx.x] += red2[threadIdx.x + s];
    }
    __syncthreads();
  }
  float mu = red1[0] / 516.f;
  float var = red2[0] / 516.f - mu * mu;
  float inv = 1.f / (sqrtf(var + EPSF) + EPSF);
  int nhb = h * 64 + b;
  for (int idx = threadIdx.x; idx < 516; idx += 128) {
    float v = (val[idx] - mu) * inv * g[idx] + gb[idx];
    if (j == 0)      Qb[((long)nhb * 4 + t) * 516 + idx] = v;
    else if (j == 1) Ktb[((long)nhb * 516 + idx) * 4 + t] = v;
    else             Vb[((long)nhb * 4 + t) * 516 + idx] = v;
  }
}

// ---------------------------------------------------------------------------
// Attention over 64 keys (60 cached + 4 new); one block per (nhb,t).
// Writes O directly in (B, NH*E=16, T, F) layout.
// ---------------------------------------------------------------------------
__global__ __launch_bounds__(128) void k_attn(
    const float* __restrict__ Qb, const float* __restrict__ Ktb,
    const float* __restrict__ Vb, const float* __restrict__ cK,
    const float* __restrict__ cV, float* __restrict__ Ob) {
  __shared__ float sc[64];
  __shared__ float att[64];
  int nhb = blockIdx.x >> 2;
  int t = blockIdx.x & 3;
  const float* Qr = Qb + ((long)nhb * 4 + t) * 516;
  if (threadIdx.x < 64) {
    int s = threadIdx.x;
    float acc = 0.f;
    if (s < 60) {
      const float* kc = cK + (long)nhb * 516 * 60 + s;
      for (int d = 0; d < 516; ++d) acc += Qr[d] * kc[(long)d * 60];
    } else {
      const float* kc = Ktb + (long)nhb * 516 * 4 + (s - 60);
      for (int d = 0; d < 516; ++d) acc += Qr[d] * kc[(long)d * 4];
    }
    sc[s] = acc * rsqrtf(516.f);
  }
  __syncthreads();
  float mx = -1e30f;
  for (int s = 0; s < 64; ++s) mx = fmaxf(mx, sc[s]);
  if (threadIdx.x < 64) att[threadIdx.x] = expf(sc[threadIdx.x] - mx);
  __syncthreads();
  float den = 0.f;
  for (int s = 0; s < 64; ++s) den += att[s];
  float iden = 1.f / den;
  int hh = nhb >> 6, bb = nhb & 63;
  for (int d = threadIdx.x; d < 516; d += 128) {
    float o = 0.f;
    for (int s = 0; s < 60; ++s) o += att[s] * cV[((long)nhb * 60 + s) * 516 + d];
    for (int s = 0; s < 4; ++s)  o += att[60 + s] * Vb[((long)nhb * 4 + s) * 516 + d];
    o *= iden;
    int e = d / 129, f = d % 129;
    Ob[(((long)bb * 16 + hh * 4 + e) * 4 + t) * 129 + f] = o;
  }
}

// new K cache: shift out 4 oldest, append Kt
__global__ void k_cacheK(const float* __restrict__ cK, const float* __restrict__ Ktb,
                         float* __restrict__ dst) {
  long idx = (long)blockIdx.x * blockDim.x + threadIdx.x;
  const long total = 256L * 516 * 60;
  if (idx >= total) return;
  int s = idx % 60;
  long nd = idx / 60;  // nhb*516 + d
  dst[idx] = (s < 56) ? cK[nd * 60 + s + 4] : Ktb[nd * 4 + (s - 56)];
}
// new V cache
__global__ void k_cacheV(const float* __restrict__ cV, const float* __restrict__ Vb,
                         float* __restrict__ dst) {
  long idx = (long)blockIdx.x * blockDim.x + threadIdx.x;
  const long total = 256L * 60 * 516;
  if (idx >= total) return;
  int d = idx % 516; long r = idx / 516;
  int s = r % 60; int nhb = r / 60;
  dst[idx] = (s < 56) ? cV[((long)nhb * 60 + s + 4) * 516 + d]
                      : Vb[((long)nhb * 4 + (s - 56)) * 516 + d];
}

// ---------------------------------------------------------------------------
// Output projection (16->32) + PReLU + ln4dcf over (C,F)=4128 + residual.
// One block per (b,t).
// ---------------------------------------------------------------------------
__global__ __launch_bounds__(128) void k_proj_ln(
    const float* __restrict__ Ob, const float* __restrict__ inter,
    const float* __restrict__ pw, const float* __restrict__ pb,
    const float* __restrict__ pa, const float* __restrict__ plg,
    const float* __restrict__ plb, float* __restrict__ esti_out) {
  __shared__ float p[4128];
  __shared__ float red1[128], red2[128];
  int b = blockIdx.x >> 2, t = blockIdx.x & 3;
  float a = pa[0];
  float s1 = 0.f, s2 = 0.f;
  for (int idx = threadIdx.x; idx < 4128; idx += 128) {
    int c = idx / 129, f = idx % 129;
    float v = pb[c];
    for (int cc = 0; cc < 16; ++cc)
      v += pw[c * 16 + cc] * Ob[((b * 16 + cc) * 4 + t) * 129 + f];
    v = (v >= 0.f) ? v : a * v;
    p[idx] = v; s1 += v; s2 += v * v;
  }
  red1[threadIdx.x] = s1; red2[threadIdx.x] = s2;
  __syncthreads();
  for (int s = 64; s > 0; s >>= 1) {
    if (threadIdx.x < s) {
      red1[threadIdx.x] += red1[threadIdx.x + s];
      red2[threadIdx.x] += red2[threadIdx.x + s];
    }
    __syncthreads();
  }
  float mu = red1[0] / 4128.f;
  float var = red2[0] / 4128.f - mu * mu;
  float inv = 1.f / (sqrtf(var + EPSF) + EPSF);
  for (int idx = threadIdx.x; idx < 4128; idx += 128) {
    int c = idx / 129, f = idx % 129;
    long xi = ((long)(b * 32 + c) * 4 + t) * 129 + f;
    esti_out[xi] = (p[idx] - mu) * inv * plg[c * 129 + f] + plb[c * 129 + f] + inter[xi];
  }
}

// ---------------------------------------------------------------------------
// Decoder: transposed-conv-equivalent 3x3 (ph=2, pw=1, flipped weights),
// t in 0..5; t<4 -> out (+cache, transposed to (B,F,T,2)); t>=4 -> dec cache.
// ---------------------------------------------------------------------------
__global__ void k_dec(const float* __restrict__ esti, const float* __restrict__ dw,
                      const float* __restrict__ cache, float* __restrict__ outp,
                      float* __restrict__ decnew) {
  int idx = blockIdx.x * blockDim.x + threadIdx.x;
  const int total = 64 * 2 * 6 * 129;
  if (idx >= total) return;
  int f = idx % 129; int r = idx / 129;
  int t = r % 6; r /= 6;
  int o = r % 2; int b = r / 2;
  float acc = 0.f;
  for (int i = 0; i < 32; ++i)
#pragma unroll
    for (int kh = 0; kh < 3; ++kh)
#pragma unroll
      for (int kw = 0; kw < 3; ++kw) {
        int tt = t + kh - 2;
        int ff = f + kw - 1;
        if (tt < 0 || tt >= 4 || ff < 0 || ff >= 129) continue;
        acc += esti[((b * 32 + i) * 4 + tt) * 129 + ff] *
               dw[((i * 2 + o) * 3 + (2 - kh)) * 3 + (2 - kw)];
      }
  if (t < 4)
    outp[(((long)b * 129 + f) * 4 + t) * 2 + o] =
        acc + cache[((b * 2 + o) * 4 + t) * 129 + f];
  else
    decnew[((b * 2 + o) * 4 + (t - 4)) * 129 + f] = acc;
}

// dec cache rows t=2,3 keep the old cache values
__global__ void k_dec_cache_copy(const float* __restrict__ cache, float* __restrict__ dst) {
  int idx = blockIdx.x * blockDim.x + threadIdx.x;
  const int total = 64 * 2 * 2 * 129;
  if (idx >= total) return;
  int f = idx % 129; int r = idx / 129;
  int tp = r % 2; r /= 2;
  int o = r % 2; int b = r / 2;
  long di = ((long)(b * 2 + o) * 4 + (2 + tp)) * 129 + f;
  dst[di] = cache[di];
}

// ---------------------------------------------------------------------------
// Host orchestration
// ---------------------------------------------------------------------------
static inline void gemm(const float* A, const float* Bm, const float* bias, float* Cm,
                        int M, int N, int K, int bT, hipStream_t s) {
  dim3 g((N + 127) / 128, (M + 63) / 64);
  if (bT) k_gemm_t<1><<<g, 256, 0, s>>>(A, Bm, bias, Cm, M, N, K);
  else    k_gemm_t<0><<<g, 256, 0, s>>>(A, Bm, bias, Cm, M, N, K);
}
static inline int gsz(long n) { return (int)((n + 255) / 256); }

extern "C" void kernel_launch(void* const* d_in, const int* in_sizes, int n_in,
                              void* d_out_v, int out_size, void* d_ws, size_t ws_size,
                              hipStream_t stream) {
  (void)in_sizes; (void)n_in; (void)out_size; (void)ws_size;
  const float* mix       = (const float*)d_in[0];
  const float* aux       = (const float*)d_in[1];
  const float* gru_h0    = (const float*)d_in[2];
  const float* attn_K    = (const float*)d_in[3];
  const float* attn_V    = (const float*)d_in[4];
  const float* enc_cache = (const float*)d_in[5];
  const float* dec_cache = (const float*)d_in[6];
  const float* enc_w     = (const float*)d_in[7];
  const float* enc_b     = (const float*)d_in[8];
  const float* bn_w      = (const float*)d_in[9];
  const float* bn_b      = (const float*)d_in[10];
  const float* bn_m      = (const float*)d_in[11];
  const float* bn_v      = (const float*)d_in[12];
  const float* fus_in_w  = (const float*)d_in[13];
  const float* fus_in_b  = (const float*)d_in[14];
  const float* fus_out_w = (const float*)d_in[15];
  const float* fus_out_b = (const float*)d_in[16];
  const float* fus_cv_w  = (const float*)d_in[17];
  const float* fus_cv_b  = (const float*)d_in[18];
  const float* intra_g   = (const float*)d_in[19];
  const float* intra_be  = (const float*)d_in[20];
  const float* intra_wih = (const float*)d_in[21];
  const float* intra_whh = (const float*)d_in[22];
  const float* intra_bih = (const float*)d_in[23];
  const float* intra_bhh = (const float*)d_in[24];
  const float* intra_lw  = (const float*)d_in[25];
  const float* intra_lb  = (const float*)d_in[26];
  const float* inter_g   = (const float*)d_in[27];
  const float* inter_be  = (const float*)d_in[28];
  const float* inter_wih = (const float*)d_in[29];
  const float* inter_whh = (const float*)d_in[30];
  const float* inter_bih = (const float*)d_in[31];
  const float* inter_bhh = (const float*)d_in[32];
  const float* inter_lw  = (const float*)d_in[33];
  const float* inter_lb  = (const float*)d_in[34];
  const float* qkv_w     = (const float*)d_in[35];
  const float* qkv_b     = (const float*)d_in[36];
  const float* qkv_a     = (const float*)d_in[37];
  const float* qkv_lg    = (const float*)d_in[38];
  const float* qkv_lb    = (const float*)d_in[39];
  const float* prj_w     = (const float*)d_in[40];
  const float* prj_b     = (const float*)d_in[41];
  const float* prj_a     = (const float*)d_in[42];
  const float* prj_lg    = (const float*)d_in[43];
  const float* prj_lb    = (const float*)d_in[44];
  const float* dec_w     = (const float*)d_in[45];

  float* out = (float*)d_out_v;
  float* ws  = (float*)d_ws;

  // workspace arena (floats)
  float* estiA   = ws + 0L;          // 1056768
  float* estiB   = ws + 1056768L;    // 1056768
  float* auxn    = ws + 2113536L;    // 1056768 (also xi)
  float* intraB  = ws + 3170304L;    // 1056768
  float* interB  = ws + 4227072L;    // 1056768
  float* xn      = ws + 5283840L;    // 1056768
  float* gin     = ws + 6340608L;    // 7995392 (also contrib)
  float* xpf     = ws + 14336000L;   // 19021824 (xpf, later xp2)
  float* xpb     = ws + 33357824L;   // 17989632 (xpb, later ys)
  float* hb      = ws + 56102912L;   // 1585152
  float* hcat    = ws + 57688064L;   // 11993088 (also lin)
  float* Qb      = ws + 69681152L;   // 528384
  float* Ktb     = ws + 70209536L;   // 528384
  float* Vb      = ws + 70737920L;   // 528384
  float* Ob      = ws + 71266304L;   // 528384

  // d_out sections (floats)
  const long DO_OUT = 0, DO_H = 66048, DO_K = 9576960, DO_V = 57131520,
             DO_ENC = 104686080, DO_DEC = 104719104;

  const int ESTI_N = 64 * 32 * 4 * 129;   // 1056768
  const int HN = 8256 * 192;              // 1585152

  // encoder + enc cache out
  k_enc<<<gsz(ESTI_N), 256, 0, stream>>>(mix, enc_cache, enc_w, enc_b,
                                         bn_w, bn_b, bn_m, bn_v, estiA);
  k_enc_cache<<<gsz(64 * 2 * 2 * 129), 256, 0, stream>>>(mix, out + DO_ENC);

  for (int l = 0; l < 6; ++l) {
    // ---- fusion block ----
    k_fusion_mha<<<8256, 32, 0, stream>>>(estiA, aux, fus_in_w + (long)l * 3072,
                                          fus_in_b + l * 96, fus_out_w + l * 1024,
                                          fus_out_b + l * 32, auxn);
    k_fusion_conv<<<gsz(ESTI_N), 256, 0, stream>>>(estiA, auxn, fus_cv_w + (long)l * 2048,
                                                   fus_cv_b + l * 32, estiB);
    // ---- intra (bidirectional GRU over frequency windows) ----
    k_ln4d<<<gsz(64 * 4 * 129), 256, 0, stream>>>(estiB, intra_g + l * 32,
                                                  intra_be + l * 32, auxn);  // auxn = xi
    k_build_gin<<<gsz(256L * 122 * 256), 256, 0, stream>>>(auxn, gin);
    gemm(gin, intra_wih + ((long)l * 2 + 0) * 576 * 256, intra_bih + (l * 2 + 0) * 576,
         xpf, 31232, 576, 256, 1, stream);
    gemm(gin, intra_wih + ((long)l * 2 + 1) * 576 * 256, intra_bih + (l * 2 + 1) * 576,
         xpb, 31232, 576, 256, 1, stream);
    // forward scan (fused WMMA gemm + gates, one launch per step)
    k_zero<<<gsz(256 * 192), 256, 0, stream>>>(hb, 256 * 192);
    for (int s = 0; s < 122; ++s) {
      k_gru_fused<<<16, 288, 0, stream>>>(hb, intra_whh + ((long)l * 2 + 0) * 576 * 192,
                                          intra_bhh + (l * 2 + 0) * 576, xpf, hcat,
                                          122, s, 122, s, 384, 0);
    }
    // backward scan (xp computed un-flipped; index reversed)
    k_zero<<<gsz(256 * 192), 256, 0, stream>>>(hb, 256 * 192);
    for (int s = 0; s < 122; ++s) {
      int pos = 121 - s;
      k_gru_fused<<<16, 288, 0, stream>>>(hb, intra_whh + ((long)l * 2 + 1) * 576 * 192,
                                          intra_bhh + (l * 2 + 1) * 576, xpb, hcat,
                                          122, pos, 122, pos, 384, 192);
    }
    // 384 -> (32,8) mixing GEMM then overlap-add + residual
    gemm(hcat, intra_lw + (long)l * 98304, nullptr, gin /*contrib*/,
         31232, 256, 384, 0, stream);
    k_unfold<<<gsz(256L * 32 * 129), 256, 0, stream>>>(gin, estiB, intra_lb + l * 32, intraB);
    // ---- inter (GRU over time) ----
    k_ln4d_to_xn<<<gsz(64 * 4 * 129), 256, 0, stream>>>(intraB, inter_g + l * 32,
                                                        inter_be + l * 32, xn);
    gemm(xn, inter_wih + (long)l * 576 * 32, inter_bih + l * 576,
         xpf /*xp2*/, 33024, 576, 32, 1, stream);
    k_copy<<<gsz(HN), 256, 0, stream>>>(gru_h0 + (long)l * HN, hb, HN);
    for (int s = 0; s < 4; ++s) {
      k_gru_fused<<<516, 288, 0, stream>>>(hb, inter_whh + (long)l * 576 * 192,
                                           inter_bhh + l * 576, xpf, xpb /*ys*/,
                                           4, s, 4, s, 192, 0);
    }
    k_copy<<<gsz(HN), 256, 0, stream>>>(hb, out + DO_H + (long)l * HN, HN);
    gemm(xpb /*ys*/, inter_lw + (long)l * 6144, inter_lb + l * 32,
         hcat /*lin*/, 33024, 32, 192, 1, stream);
    k_add_lin<<<gsz(ESTI_N), 256, 0, stream>>>(intraB, hcat, interB);
    // ---- frequency attention with KV cache ----
    k_qkv_make<<<3072, 128, 0, stream>>>(interB, qkv_w + (long)l * 1536,
                                         qkv_b + l * 48, qkv_a + l * 12,
                                         qkv_lg + (long)l * 6192, qkv_lb + (long)l * 6192,
                                         Qb, Ktb, Vb);
    const float* cK = attn_K + (long)l * 7925760;
    const float* cV = attn_V + (long)l * 7925760;
    k_attn<<<1024, 128, 0, stream>>>(Qb, Ktb, Vb, cK, cV, Ob);
    k_cacheK<<<gsz(7925760L), 256, 0, stream>>>(cK, Ktb, out + DO_K + (long)l * 7925760);
    k_cacheV<<<gsz(7925760L), 256, 0, stream>>>(cV, Vb, out + DO_V + (long)l * 7925760);
    k_proj_ln<<<256, 128, 0, stream>>>(Ob, interB, prj_w + (long)l * 512,
                                       prj_b + l * 32, prj_a + l,
                                       prj_lg + (long)l * 4128, prj_lb + (long)l * 4128,
                                       estiA);
  }

  // decoder + caches
  k_dec<<<gsz(64 * 2 * 6 * 129), 256, 0, stream>>>(estiA, dec_w, dec_cache,
                                                   out + DO_OUT, out + DO_DEC);
  k_dec_cache_copy<<<gsz(64 * 2 * 2 * 129), 256, 0, stream>>>(dec_cache, out + DO_DEC);
}